// LightPollutionGNN_23038204576048
// MI455X (gfx1250) — compile-verified
//
#include <hip/hip_runtime.h>
#include <hip/hip_bf16.h>

#define TPB 256

typedef float v2f __attribute__((ext_vector_type(2)));
typedef float v8f __attribute__((ext_vector_type(8)));

// ---- ordered-float <-> uint encoding for atomicMax segment-max ----
__device__ __forceinline__ unsigned fenc(float f) {
    unsigned u = __float_as_uint(f);
    return (u & 0x80000000u) ? ~u : (u | 0x80000000u);
}
__device__ __forceinline__ float fdec(unsigned u) {
    unsigned b = (u & 0x80000000u) ? (u ^ 0x80000000u) : ~u;
    return __uint_as_float(b);
}

// ---------------- utility ----------------
__global__ void k_zero(float* __restrict__ p, long long n) {
    long long t = (long long)blockIdx.x * blockDim.x + threadIdx.x;
    if (t < n) p[t] = 0.0f;
}

// x[n, 0:16] = embedding[gid[n]]; x[n,16:20] = ord_feats[n]; x[n,20:32] = 0
__global__ void k_node_feat(const int* __restrict__ gid, const float* __restrict__ ordf,
                            const float* __restrict__ emb, float* __restrict__ x, int N) {
    long long t = (long long)blockIdx.x * blockDim.x + threadIdx.x;
    if (t >= (long long)N * 32) return;
    int n = (int)(t >> 5), c = (int)(t & 31);
    float v = 0.0f;
    if (c < 16)       v = emb[(long long)gid[n] * 16 + c];
    else if (c < 20)  v = ordf[(long long)n * 4 + (c - 16)];
    x[t] = v;
}

// degree + edge_attr sums per dst (for self-loop mean fill)
__global__ void k_deg(const int* __restrict__ dst, const float* __restrict__ eattr,
                      float* __restrict__ cnt, float* __restrict__ msum, int E) {
    int t = blockIdx.x * blockDim.x + threadIdx.x;
    if (t >= E) return;
    int d = dst[t];
    atomicAdd(&cnt[d], 1.0f);
    atomicAdd(&msum[(long long)d * 2 + 0], eattr[(long long)t * 2 + 0]);
    atomicAdd(&msum[(long long)d * 2 + 1], eattr[(long long)t * 2 + 1]);
}

__global__ void k_mean(float* __restrict__ msum, const float* __restrict__ cnt, int N) {
    int t = blockIdx.x * blockDim.x + threadIdx.x;
    if (t >= N * 2) return;
    msum[t] = msum[t] / fmaxf(cnt[t >> 1], 1.0f);
}

// pad lin_w (K x 32, K<=32) into 32x32 with zero rows
__global__ void k_pack_w(const float* __restrict__ w, float* __restrict__ wpad, int K) {
    int t = blockIdx.x * blockDim.x + threadIdx.x;
    if (t >= 1024) return;
    wpad[t] = ((t >> 5) < K) ? w[t] : 0.0f;
}

// we[h*2+j] = sum_c edge_w[j,h*16+c] * att_edge[h,c]
__global__ void k_edge_coef(const float* __restrict__ edge_w, const float* __restrict__ att_edge,
                            float* __restrict__ we) {
    int t = threadIdx.x;
    if (t >= 4) return;
    int h = t >> 1, j = t & 1;
    float s = 0.0f;
    for (int c = 0; c < 16; ++c) s += edge_w[j * 32 + h * 16 + c] * att_edge[h * 16 + c];
    we[t] = s;
}

// ---------------- WMMA f32 GEMM: Y[rows x 32] = X[rows x 32] @ W[32 x 32] ----------------
// One wave per 16-row tile; two 16x16 N-tiles; K=32 as 8 steps of V_WMMA_F32_16X16X4_F32.
__global__ void k_gemm32(const float* __restrict__ X, const float* __restrict__ W,
                         float* __restrict__ Y, int rows) {
    int wave  = blockIdx.x * (TPB / 32) + (threadIdx.x >> 5);
    int lane  = threadIdx.x & 31;
    int tiles = (rows + 15) >> 4;
    if (wave >= tiles) return;               // uniform per-wave exit (EXEC stays all-1 for WMMA)
    int m0 = wave << 4;
    int col = lane & 15, half = lane >> 4;
    int rowL = m0 + col; if (rowL > rows - 1) rowL = rows - 1;   // clamped load, guarded store

    v2f a[8], b0[8], b1[8];
    for (int k = 0; k < 8; ++k) {
        int k4 = 4 * k + 2 * half;           // this lane-half's K offsets: k4, k4+1
        a[k].x = X[(long long)rowL * 32 + k4];
        a[k].y = X[(long long)rowL * 32 + k4 + 1];
        b0[k].x = W[k4 * 32 + col];
        b0[k].y = W[(k4 + 1) * 32 + col];
        b1[k].x = W[k4 * 32 + 16 + col];
        b1[k].y = W[(k4 + 1) * 32 + 16 + col];
    }
    v8f c0 = {0,0,0,0,0,0,0,0};
    v8f c1 = {0,0,0,0,0,0,0,0};
    for (int k = 0; k < 8; ++k)
        c0 = __builtin_amdgcn_wmma_f32_16x16x4_f32(false, a[k], false, b0[k], (short)0, c0, false, false);
    for (int k = 0; k < 8; ++k)
        c1 = __builtin_amdgcn_wmma_f32_16x16x4_f32(false, a[k], false, b1[k], (short)0, c1, false, false);
    for (int j = 0; j < 8; ++j) {
        int r = m0 + j + 8 * half;
        if (r < rows) {
            Y[(long long)r * 32 + col]      = c0[j];
            Y[(long long)r * 32 + 16 + col] = c1[j];
        }
    }
}

// a_src[n,h], a_dst[n,h] = <xp[n,h,:], att_{src,dst}[h,:]>
__global__ void k_attn(const float* __restrict__ xp, const float* __restrict__ att_src,
                       const float* __restrict__ att_dst, float* __restrict__ a_src,
                       float* __restrict__ a_dst, int N) {
    int t = blockIdx.x * blockDim.x + threadIdx.x;
    if (t >= N * 2) return;
    int n = t >> 1, h = t & 1;
    const float* row = xp + (long long)n * 32 + h * 16;
    float s = 0.0f, d = 0.0f;
    for (int c = 0; c < 16; ++c) {
        float v = row[c];
        s += v * att_src[h * 16 + c];
        d += v * att_dst[h * 16 + c];
    }
    a_src[t] = s;
    a_dst[t] = d;
}

// pass 1: raw leaky-relu logits + segment max via ordered-uint atomicMax
__global__ void k_alpha_max(const int* __restrict__ src, const int* __restrict__ dst,
                            const float* __restrict__ eattr, const float* __restrict__ mean_attr,
                            const float* __restrict__ a_src, const float* __restrict__ a_dst,
                            const float* __restrict__ we, float* __restrict__ alpha,
                            unsigned* __restrict__ m_enc, int E, int N) {
    long long t = (long long)blockIdx.x * blockDim.x + threadIdx.x;
    if (t >= (long long)E + N) return;
    int s, d; float ea0, ea1;
    if (t < E) {
        s = src[t]; d = dst[t];
        ea0 = eattr[t * 2]; ea1 = eattr[t * 2 + 1];
    } else {
        int n = (int)(t - E); s = n; d = n;
        ea0 = mean_attr[n * 2]; ea1 = mean_attr[n * 2 + 1];
    }
    for (int h = 0; h < 2; ++h) {
        float r = a_src[(long long)s * 2 + h] + a_dst[(long long)d * 2 + h]
                + ea0 * we[h * 2] + ea1 * we[h * 2 + 1];
        r = (r > 0.0f) ? r : 0.2f * r;
        alpha[t * 2 + h] = r;
        atomicMax(&m_enc[(long long)d * 2 + h], fenc(r));
    }
}

// pass 2: alpha = exp(raw - max); denom accumulation
__global__ void k_alpha_exp(const int* __restrict__ dst, float* __restrict__ alpha,
                            const unsigned* __restrict__ m_enc, float* __restrict__ denom,
                            int E, int N) {
    long long t = (long long)blockIdx.x * blockDim.x + threadIdx.x;
    if (t >= (long long)E + N) return;
    int d = (t < E) ? dst[t] : (int)(t - E);
    for (int h = 0; h < 2; ++h) {
        float a = expf(alpha[t * 2 + h] - fdec(m_enc[(long long)d * 2 + h]));
        alpha[t * 2 + h] = a;
        atomicAdd(&denom[(long long)d * 2 + h], a);
    }
}

// pass 3: message scatter; one wave per edge, lane = channel (coalesced 128B gather/scatter)
__global__ void k_message(const int* __restrict__ src, const int* __restrict__ dst,
                          const float* __restrict__ xp, const float* __restrict__ alpha,
                          const float* __restrict__ denom, float* __restrict__ outacc,
                          int E, int N) {
    long long t = (long long)blockIdx.x * blockDim.x + threadIdx.x;
    long long EN = (long long)E + N;
    if (t >= EN * 32) return;
    long long e = t >> 5;
    int c = (int)(t & 31), h = c >> 4;
    int s, d;
    if (e < E) { s = src[e]; d = dst[e]; }
    else       { s = d = (int)(e - E); }
    float w = alpha[e * 2 + h] / (denom[(long long)d * 2 + h] + 1e-16f);
    atomicAdd(&outacc[(long long)d * 32 + c], w * xp[(long long)s * 32 + c]);
}

__global__ void k_relu_bias(const float* __restrict__ acc, const float* __restrict__ bias,
                            float* __restrict__ x, int N) {
    long long t = (long long)blockIdx.x * blockDim.x + threadIdx.x;
    if (t >= (long long)N * 32) return;
    x[t] = fmaxf(acc[t] + bias[t & 31], 0.0f);
}

// sparse pooling: wave per nnz, lane per channel
__global__ void k_pool(const int* __restrict__ prow, const int* __restrict__ pcol,
                       const float* __restrict__ pval, const float* __restrict__ x,
                       float* __restrict__ pooled, long long NNZ) {
    long long t = (long long)blockIdx.x * blockDim.x + threadIdx.x;
    if (t >= NNZ * 32) return;
    long long i = t >> 5;
    int c = (int)(t & 31);
    atomicAdd(&pooled[(long long)prow[i] * 32 + c], pval[i] * x[(long long)pcol[i] * 32 + c]);
}

// head: hh = relu(pooled @ W1 + b1) via WMMA; out = hh @ W2 + b2 via shfl reduce
__global__ void k_head(const float* __restrict__ pooled, const float* __restrict__ W1,
                       const float* __restrict__ b1, const float* __restrict__ W2,
                       const float* __restrict__ b2, float* __restrict__ out, int P) {
    int wave  = blockIdx.x * (TPB / 32) + (threadIdx.x >> 5);
    int lane  = threadIdx.x & 31;
    int tiles = (P + 15) >> 4;
    if (wave >= tiles) return;
    int m0 = wave << 4;
    int col = lane & 15, half = lane >> 4;
    int rowL = m0 + col; if (rowL > P - 1) rowL = P - 1;

    v2f a[8], b0[8], b1v[8];
    for (int k = 0; k < 8; ++k) {
        int k4 = 4 * k + 2 * half;
        a[k].x  = pooled[(long long)rowL * 32 + k4];
        a[k].y  = pooled[(long long)rowL * 32 + k4 + 1];
        b0[k].x  = W1[k4 * 32 + col];
        b0[k].y  = W1[(k4 + 1) * 32 + col];
        b1v[k].x = W1[k4 * 32 + 16 + col];
        b1v[k].y = W1[(k4 + 1) * 32 + 16 + col];
    }
    v8f c0 = {0,0,0,0,0,0,0,0};
    v8f c1 = {0,0,0,0,0,0,0,0};
    for (int k = 0; k < 8; ++k)
        c0 = __builtin_amdgcn_wmma_f32_16x16x4_f32(false, a[k], false, b0[k],  (short)0, c0, false, false);
    for (int k = 0; k < 8; ++k)
        c1 = __builtin_amdgcn_wmma_f32_16x16x4_f32(false, a[k], false, b1v[k], (short)0, c1, false, false);

    float w2lo = W2[col], w2hi = W2[16 + col];
    float bb0 = b1[col],  bb1 = b1[16 + col];
    float part[8];
    for (int j = 0; j < 8; ++j)
        part[j] = fmaxf(c0[j] + bb0, 0.0f) * w2lo + fmaxf(c1[j] + bb1, 0.0f) * w2hi;
    // reduce over the 16 lanes of each half (D rows split at lane 16 in wave32)
    for (int off = 1; off < 16; off <<= 1)
        for (int j = 0; j < 8; ++j)
            part[j] += __shfl_xor(part[j], off, 32);
    if (col == 0) {
        float bb2 = b2[0];
        for (int j = 0; j < 8; ++j) {
            int r = m0 + j + 8 * half;
            if (r < P) out[r] = part[j] + bb2;
        }
    }
}

// ---------------- launch ----------------
static inline unsigned nblk(long long n) { return (unsigned)((n + TPB - 1) / TPB); }

extern "C" void kernel_launch(void* const* d_in, const int* in_sizes, int n_in,
                              void* d_out, int out_size, void* d_ws, size_t ws_size,
                              hipStream_t stream) {
    const int*   group_ids = (const int*)d_in[0];
    const float* ord_feats = (const float*)d_in[1];
    const int*   edge_index= (const int*)d_in[2];
    const float* edge_attr = (const float*)d_in[3];
    const int*   pool_rows = (const int*)d_in[4];
    const int*   pool_cols = (const int*)d_in[5];
    const float* pool_vals = (const float*)d_in[6];
    const float* embedding = (const float*)d_in[8];
    const float* lin_w[3]  = {(const float*)d_in[9],  (const float*)d_in[15], (const float*)d_in[21]};
    const float* att_s[3]  = {(const float*)d_in[10], (const float*)d_in[16], (const float*)d_in[22]};
    const float* att_d[3]  = {(const float*)d_in[11], (const float*)d_in[17], (const float*)d_in[23]};
    const float* edge_w[3] = {(const float*)d_in[12], (const float*)d_in[18], (const float*)d_in[24]};
    const float* att_e[3]  = {(const float*)d_in[13], (const float*)d_in[19], (const float*)d_in[25]};
    const float* bias[3]   = {(const float*)d_in[14], (const float*)d_in[20], (const float*)d_in[26]};
    const float* head_w1 = (const float*)d_in[27];
    const float* head_b1 = (const float*)d_in[28];
    const float* head_w2 = (const float*)d_in[29];
    const float* head_b2 = (const float*)d_in[30];

    const int N = in_sizes[0];
    const int E = in_sizes[3] / 2;
    const long long NNZ = in_sizes[4];
    const int P = out_size;
    const long long EN = (long long)E + N;

    // workspace layout (floats)
    float* x       = (float*)d_ws;
    float* xp      = x      + (long long)N * 32;
    float* outacc  = xp     + (long long)N * 32;
    float* alpha   = outacc + (long long)N * 32;   // EN*2
    float* a_src   = alpha  + EN * 2;              // N*2
    float* a_dst   = a_src  + (long long)N * 2;
    float* denom   = a_dst  + (long long)N * 2;    // N*2 (m_enc follows: zero both at once)
    unsigned* m_enc= (unsigned*)(denom + (long long)N * 2);  // N*2
    float* meanat  = (float*)(m_enc + (long long)N * 2);     // N*2
    float* cnt     = meanat + (long long)N * 2;              // N
    float* wpad    = cnt    + N;                             // 1024
    float* we      = wpad   + 1024;                          // 4
    float* pooled  = we     + 4;                             // P*32

    const int* src = edge_index;
    const int* dst = edge_index + E;

    // node features + self-loop mean edge_attr
    k_node_feat<<<nblk((long long)N * 32), TPB, 0, stream>>>(group_ids, ord_feats, embedding, x, N);
    k_zero<<<nblk((long long)N * 3), TPB, 0, stream>>>(meanat, (long long)N * 3); // meanat + cnt
    k_deg<<<nblk(E), TPB, 0, stream>>>(dst, edge_attr, cnt, meanat, E);
    k_mean<<<nblk((long long)N * 2), TPB, 0, stream>>>(meanat, cnt, N);

    const int gemm_blocks = (((N + 15) / 16) + (TPB / 32) - 1) / (TPB / 32);

    for (int l = 0; l < 3; ++l) {
        const int K = (l == 0) ? 20 : 32;
        k_pack_w<<<4, TPB, 0, stream>>>(lin_w[l], wpad, K);
        k_edge_coef<<<1, 4, 0, stream>>>(edge_w[l], att_e[l], we);
        k_gemm32<<<gemm_blocks, TPB, 0, stream>>>(x, wpad, xp, N);
        k_attn<<<nblk((long long)N * 2), TPB, 0, stream>>>(xp, att_s[l], att_d[l], a_src, a_dst, N);
        k_zero<<<nblk((long long)N * 32), TPB, 0, stream>>>(outacc, (long long)N * 32);
        k_zero<<<nblk((long long)N * 4), TPB, 0, stream>>>(denom, (long long)N * 4); // denom + m_enc
        k_alpha_max<<<nblk(EN), TPB, 0, stream>>>(src, dst, edge_attr, meanat, a_src, a_dst,
                                                  we, alpha, m_enc, E, N);
        k_alpha_exp<<<nblk(EN), TPB, 0, stream>>>(dst, alpha, m_enc, denom, E, N);
        k_message<<<nblk(EN * 32), TPB, 0, stream>>>(src, dst, xp, alpha, denom, outacc, E, N);
        k_relu_bias<<<nblk((long long)N * 32), TPB, 0, stream>>>(outacc, bias[l], x, N);
    }

    // pooling + head
    k_zero<<<nblk((long long)P * 32), TPB, 0, stream>>>(pooled, (long long)P * 32);
    k_pool<<<nblk(NNZ * 32), TPB, 0, stream>>>(pool_rows, pool_cols, pool_vals, x, pooled, NNZ);
    const int head_blocks = (((P + 15) / 16) + (TPB / 32) - 1) / (TPB / 32);
    k_head<<<head_blocks, TPB, 0, stream>>>(pooled, head_w1, head_b1, head_w2, head_b2,
                                            (float*)d_out, P);
}